// Our_31456340476445
// MI455X (gfx1250) — compile-verified
//
#include <hip/hip_runtime.h>

// ---------------- problem constants (match reference) ----------------
#define U_N   100000
#define I_N   50000
#define D_DIM 64
#define H_N   4
#define DH_N  16
#define HN_N  128
#define L_N   2
#define NNZ   1000000
#define SLOPE 0.5f

typedef float v2f __attribute__((ext_vector_type(2)));
typedef float v8f __attribute__((ext_vector_type(8)));

// TDM descriptor operand types (amdgpu-toolchain / clang-23 6-arg form:
//  (uint32x4 g0, int32x8 g1, int32x4, int32x4, int32x8, i32 cpol))
typedef unsigned int u32x4 __attribute__((ext_vector_type(4)));
typedef int          i32x8 __attribute__((ext_vector_type(8)));
typedef int          i32x4 __attribute__((ext_vector_type(4)));

#if defined(__has_builtin)
#if __has_builtin(__builtin_amdgcn_tensor_load_to_lds) && \
    __has_builtin(__builtin_amdgcn_s_wait_tensorcnt)
#define HAVE_TDM 1
#endif
#endif

static inline int cdiv(long long a, long long b) { return (int)((a + b - 1) / b); }

// ---------------- elementwise utility kernels ----------------
__global__ void k_zero(float* __restrict__ p, int n) {
    int i = blockIdx.x * 256 + threadIdx.x;
    if (i < n) p[i] = 0.f;
}
__global__ void k_copy(float* __restrict__ o, const float* __restrict__ a, int n) {
    int i = blockIdx.x * 256 + threadIdx.x;
    if (i < n) o[i] = a[i];
}
__global__ void k_add3(float* __restrict__ o, const float* __restrict__ a,
                       const float* __restrict__ b, const float* __restrict__ c, int n) {
    int i = blockIdx.x * 256 + threadIdx.x;
    if (i < n) o[i] = a[i] + b[i] + c[i];
}

// ---------------- COO SpMM: Y[r,:] += v * X[c,:], D=64, atomic scatter ----------------
__global__ __launch_bounds__(256) void k_spmm(const int* __restrict__ rows,
                                              const int* __restrict__ cols,
                                              const float* __restrict__ vals,
                                              const float* __restrict__ X,
                                              float* __restrict__ Y, int nnz) {
    int tid = blockIdx.x * 256 + threadIdx.x;
    int e = tid >> 4;
    if (e >= nnz) return;
    int q = tid & 15;
    int r = rows[e], c = cols[e];
    float v = vals[e];
    float4 x = ((const float4*)X)[c * 16 + q];
    float* y = Y + ((long long)r * 64 + q * 4);
    atomicAdd(y + 0, v * x.x);
    atomicAdd(y + 1, v * x.y);
    atomicAdd(y + 2, v * x.z);
    atomicAdd(y + 3, v * x.w);
}

// ---------------- WMMA GEMM: C[N,64] = A[N,64] @ B[64,64] (fp32, 16x16x4) ----------------
// keymode==0: out row-major [N,64].  keymode==1: out[(h*N + row)*16 + col16]  (Key layout [H,N,DH]).
// 8 waves/block, one 16-row tile each. A tiles DMA'd to LDS by the Tensor Data Mover.
// B staged into a row-pair-interleaved LDS image so each lane's {k,k+1} B pair is one ds_load_b64.
__global__ __launch_bounds__(256) void k_gemm64(const float* __restrict__ A,
                                                const float* __restrict__ B,
                                                float* __restrict__ out,
                                                int ntiles, int N, int keymode) {
    __shared__ float ldsB2[64 * 64];      // pair p=k/2: ldsB2[p*128 + col*2 + (k&1)]
    __shared__ float ldsA[8][16 * 64];    // row-major per-wave tile
    int tid  = threadIdx.x;
    int wave = tid >> 5;
    int lane = tid & 31;
    int tile = blockIdx.x * 8 + wave;
    bool active = (tile < ntiles);        // wave-uniform

#ifdef HAVE_TDM
    {   // Tensor Data Mover: DMA this wave's 16x64 fp32 tile (4KB contiguous) into LDS.
        unsigned long long gaddr =
            (unsigned long long)(const void*)(A + (long long)tile * 1024) * (active ? 1u : 0u);
        unsigned int ldsa = (unsigned int)(unsigned long long)(void*)&ldsA[wave][0];
        u32x4 g0;
        g0[0] = active ? 1u : 0u;                                   // count=1 (valid) / 0 (NULL)
        g0[1] = ldsa;                                               // lds_addr
        g0[2] = (unsigned int)gaddr;                                // global_addr[31:0]
        g0[3] = ((unsigned int)(gaddr >> 32) & 0x01FFFFFFu) | 0x80000000u; // addr[56:32] | type=2
        // group1: data_size=4B (code 2); 1-row tile of 1024 elements.
        unsigned long long td0 = 1024, td1 = 1, tl0 = 1024, tl1 = 1, tl2 = 0;
        unsigned long long st0 = 1024, st1 = 1024;
        unsigned long long q0 = (2ull << 16) | ((td0 & 0xFFFFull) << 48);
        unsigned long long q1 = (td0 >> 16) | (td1 << 16) | (tl0 << 48);
        unsigned long long q2 = tl1 | (tl2 << 16) | ((st0 & 0xFFFFFFFFull) << 32);
        unsigned long long q3 = (st0 >> 32) | (st1 << 16);
        union { unsigned long long q[4]; i32x8 v; } g1u;
        g1u.q[0] = q0; g1u.q[1] = q1; g1u.q[2] = q2; g1u.q[3] = q3;
        i32x4 gz4 = {0, 0, 0, 0};
        i32x8 gz8 = {0, 0, 0, 0, 0, 0, 0, 0};
        __builtin_amdgcn_tensor_load_to_lds(g0, g1u.v, gz4, gz4, gz8, 0);
    }
#else
    if (active) {
        const float* At = A + (long long)tile * 1024;
        for (int idx = lane; idx < 16 * 16; idx += 32)
            ((float4*)ldsA[wave])[idx] = ((const float4*)At)[idx];
    }
#endif

    // stage B into pair-interleaved layout (all 256 threads)
    for (int idx = tid; idx < 64 * 64; idx += 256) {
        int k = idx >> 6, c = idx & 63;
        ldsB2[(k >> 1) * 128 + c * 2 + (k & 1)] = B[idx];
    }

#ifdef HAVE_TDM
    __builtin_amdgcn_s_wait_tensorcnt(0);   // this wave's A tile resident before barrier
#endif
    __syncthreads();
    if (!active) return;

    int r16 = lane & 15;
    int hi  = lane >> 4;     // half-wave selects the K pair (ISA 16x4-A / 4x16-B layout)
    v8f acc[4] = {};
    const float* la = ldsA[wave];

    for (int kk = 0; kk < 64; kk += 4) {
        v2f a = *(const v2f*)&la[r16 * 64 + kk + 2 * hi];          // A[row][k..k+1], 8B aligned
        const float* bp = &ldsB2[(kk / 2 + hi) * 128];
#pragma unroll
        for (int c = 0; c < 4; ++c) {
            v2f b = *(const v2f*)&bp[(c * 16 + r16) * 2];          // B pair, one ds_load_b64
            acc[c] = __builtin_amdgcn_wmma_f32_16x16x4_f32(false, a, false, b,
                                                           (short)0, acc[c], false, false);
        }
    }
#pragma unroll
    for (int c = 0; c < 4; ++c) {
#pragma unroll
        for (int j = 0; j < 8; ++j) {
            int row = tile * 16 + j + 8 * hi;
            if (keymode) out[((long long)c * N + row) * 16 + r16] = acc[c][j];
            else         out[(long long)row * 64 + c * 16 + r16] = acc[c][j];
        }
    }
}

// ---------------- WMMA A^T B reduction: VK[h] += Vl_head^T @ Key_head over N rows ----------------
// Pair-interleaved LDS staging: rows 2p,2p+1 interleaved so each fragment is one ds_load_b64.
__global__ __launch_bounds__(32) void k_atb(const float* __restrict__ Vl,
                                            const float* __restrict__ Key,
                                            float* __restrict__ VK, int N) {
    __shared__ float lA2[64 * 16];   // lA2[(n/2)*32 + i*2 + (n&1)]
    __shared__ float lB2[64 * 16];
    int lane = threadIdx.x;
    int h    = blockIdx.y;
    int n0   = blockIdx.x * 64;

    for (int idx = lane; idx < 64 * 16; idx += 32) {
        int n = idx >> 4;
        int i = idx & 15;
        int gn = n0 + n;
        int p  = (n >> 1) * 32 + i * 2 + (n & 1);
        lA2[p] = (gn < N) ? Vl[(long long)gn * 64 + h * 16 + i] : 0.f;
        lB2[p] = (gn < N) ? Key[((long long)h * N + gn) * 16 + i] : 0.f;
    }
    __syncthreads();

    int r16 = lane & 15, hi = lane >> 4;
    v8f acc = {};
    for (int kk = 0; kk < 64; kk += 4) {              // K dimension = rows n
        int p = (kk / 2 + hi) * 32 + r16 * 2;
        v2f a = *(const v2f*)&lA2[p];                 // A[M=i, K=n..n+1]
        v2f b = *(const v2f*)&lB2[p];                 // B[K=n..n+1, N=j]
        acc = __builtin_amdgcn_wmma_f32_16x16x4_f32(false, a, false, b,
                                                    (short)0, acc, false, false);
    }
    float* vkh = VK + h * 256;
#pragma unroll
    for (int j = 0; j < 8; ++j)
        atomicAdd(&vkh[(j + 8 * hi) * 16 + r16], acc[j]);
}

// ---------------- WMMA per-head expansion: new = Key[h] @ pre2[h]; cur=new; total+=new ----------------
__global__ __launch_bounds__(32) void k_key_pre2(const float* __restrict__ Key,
                                                 const float* __restrict__ pre2,
                                                 float* __restrict__ cur,
                                                 float* __restrict__ total, int N) {
    __shared__ float lK[16 * 16];    // row-major: A pairs are contiguous
    __shared__ float lP2[16 * 16];   // pair-interleaved pre2[h]
    int lane = threadIdx.x;
    int h    = blockIdx.y;
    int n0   = blockIdx.x * 16;

    for (int idx = lane; idx < 256; idx += 32) {
        lK[idx] = Key[((long long)h * N + n0) * 16 + idx];
        int d = idx >> 4, j = idx & 15;
        lP2[(d >> 1) * 32 + j * 2 + (d & 1)] = pre2[h * 256 + idx];
    }
    __syncthreads();

    int r16 = lane & 15, hi = lane >> 4;
    v8f acc = {};
    for (int kk = 0; kk < 16; kk += 4) {
        v2f a = *(const v2f*)&lK[r16 * 16 + kk + 2 * hi];        // A[M=n', K=d..d+1]
        v2f b = *(const v2f*)&lP2[(kk / 2 + hi) * 32 + r16 * 2]; // B[K=d..d+1, N=j]
        acc = __builtin_amdgcn_wmma_f32_16x16x4_f32(false, a, false, b,
                                                    (short)0, acc, false, false);
    }
#pragma unroll
    for (int j = 0; j < 8; ++j) {
        long long row = n0 + j + 8 * hi;
        int col = h * 16 + r16;
        float v = acc[j];
        cur[row * 64 + col] = v;
        total[row * 64 + col] += v;
    }
}

// ---------------- tiny dense chain (negligible flops -> plain VALU) ----------------
__device__ __forceinline__ float leakyf(float x) { return x >= 0.f ? x : SLOPE * x; }

__global__ void k_t1(const float* __restrict__ VK, const float* __restrict__ Hyper,
                     float* __restrict__ t1) {
    int idx = blockIdx.x * 256 + threadIdx.x;
    if (idx >= 64 * 128) return;
    int r = idx >> 7, m = idx & 127;
    int h = r >> 4, i = r & 15;
    float s = 0.f;
    for (int j = 0; j < 16; ++j)
        s += VK[h * 256 + i * 16 + j] * Hyper[m * 64 + h * 16 + j];
    t1[r * 128 + m] = s;
}

__global__ void k_reslin(const float* __restrict__ tin, const float* __restrict__ W,
                         float* __restrict__ tout) {
    int idx = blockIdx.x * 256 + threadIdx.x;
    if (idx >= 64 * 128) return;
    int r = idx >> 7, m = idx & 127;
    float s = 0.f;
    for (int k = 0; k < 128; ++k) s += tin[r * 128 + k] * W[m * 128 + k];
    tout[r * 128 + m] = leakyf(s) + tin[r * 128 + m];
}

__global__ void k_P(const float* __restrict__ t3, const float* __restrict__ V,
                    float* __restrict__ P) {
    int idx = blockIdx.x * 256 + threadIdx.x;
    if (idx >= 128 * 64) return;
    int m = idx >> 6, c = idx & 63;
    float s = 0.f;
    for (int r = 0; r < 64; ++r) s += t3[r * 128 + m] * V[r * 64 + c];
    P[m * 64 + c] = s;
}

__global__ void k_pre2(const float* __restrict__ Hyper, const float* __restrict__ P,
                       float* __restrict__ pre2) {
    int idx = blockIdx.x * 256 + threadIdx.x;
    if (idx >= 1024) return;
    int h = idx >> 8, i = (idx >> 4) & 15, j = idx & 15;
    float s = 0.f;
    for (int m = 0; m < 128; ++m)
        s += Hyper[m * 64 + h * 16 + i] * P[m * 64 + h * 16 + j];
    pre2[idx] = s;
}

// ---------------- host-side transformer driver ----------------
static void run_transformer(const float* E0, const float* Key, const float* Hyper,
                            const float* V, const float* pV,
                            const float* W1base, const float* W2base,
                            float* total, float* hyperOut,
                            float* bufCur, float* bufVl, float* sm,
                            int N, hipStream_t stream) {
    float* VK   = sm;
    float* t1   = VK + 1024;
    float* t2   = t1 + 64 * 128;
    float* t3   = t2 + 64 * 128;
    float* P    = t3 + 64 * 128;
    float* pre2 = P + 128 * 64;

    int nE = N * 64;
    k_copy<<<cdiv(nE, 256), 256, 0, stream>>>(total, E0, nE);
    k_copy<<<cdiv(HN_N * 64, 256), 256, 0, stream>>>(hyperOut, Hyper, HN_N * 64);

    int ntiles = N / 16;
    const float* cur = E0;
    for (int l = 0; l < L_N; ++l) {
        k_gemm64<<<cdiv(ntiles, 8), 256, 0, stream>>>(cur, pV, bufVl, ntiles, N, 0);
        k_zero<<<cdiv(1024, 256), 256, 0, stream>>>(VK, 1024);
        dim3 gatb(cdiv(N, 64), H_N);
        k_atb<<<gatb, 32, 0, stream>>>(bufVl, Key, VK, N);
        k_t1<<<cdiv(64 * 128, 256), 256, 0, stream>>>(VK, Hyper, t1);
        k_reslin<<<cdiv(64 * 128, 256), 256, 0, stream>>>(t1, W1base + (long long)l * 128 * 128, t2);
        k_reslin<<<cdiv(64 * 128, 256), 256, 0, stream>>>(t2, W2base + (long long)l * 128 * 128, t3);
        k_P<<<cdiv(128 * 64, 256), 256, 0, stream>>>(t3, V, P);
        k_pre2<<<cdiv(1024, 256), 256, 0, stream>>>(Hyper, P, pre2);
        dim3 gkp(ntiles, H_N);
        k_key_pre2<<<gkp, 32, 0, stream>>>(Key, pre2, bufCur, total, N);
        cur = bufCur;
    }
}

// ---------------- entry point ----------------
extern "C" void kernel_launch(void* const* d_in, const int* in_sizes, int n_in,
                              void* d_out, int out_size, void* d_ws, size_t ws_size,
                              hipStream_t stream) {
    const float* uE    = (const float*)d_in[0];
    const float* iE    = (const float*)d_in[1];
    const float* Ks    = (const float*)d_in[2];
    const float* Hyp   = (const float*)d_in[3];
    const float* hgV   = (const float*)d_in[4];
    const float* hgpV  = (const float*)d_in[5];
    const float* hgW1  = (const float*)d_in[6];
    const float* hgW2  = (const float*)d_in[7];
    const int*   a_r   = (const int*)d_in[8];
    const int*   a_c   = (const int*)d_in[9];
    const float* a_v   = (const float*)d_in[10];
    const int*   t_r   = (const int*)d_in[11];
    const int*   t_c   = (const int*)d_in[12];
    const float* t_v   = (const float*)d_in[13];
    const int*   u_r   = (const int*)d_in[14];
    const int*   u_c   = (const int*)d_in[15];
    const float* u_v   = (const float*)d_in[16];

    const long long U64 = (long long)U_N * 64;
    const long long I64 = (long long)I_N * 64;

    float* out = (float*)d_out;
    float* o_uiU    = out;
    float* o_uiI    = o_uiU + U64;
    float* o_uiUlat = o_uiI + I64;
    float* o_uiIlat = o_uiUlat + U64;
    float* o_uiUKey = o_uiIlat + I64;
    float* o_uiIKey = o_uiUKey + U64;
    float* o_uiUHyp = o_uiIKey + I64;
    float* o_uiIHyp = o_uiUHyp + HN_N * 64;
    float* o_uuE    = o_uiIHyp + HN_N * 64;
    float* o_uuLat  = o_uuE + U64;
    float* o_uuKey  = o_uuLat + U64;
    float* o_uuHyp  = o_uuKey + U64;

    float* ws   = (float*)d_ws;
    float* bufA = ws;
    float* bufB = bufA + U64;
    float* bufC = bufB + U64;
    float* bufD = bufC + I64;
    float* sm   = bufD + I64;

    const int spmmGrid = cdiv((long long)NNZ * 16, 256);

    // -------- lightgcn (ui bipartite, 2 hops) --------
    k_zero<<<cdiv(U64, 256), 256, 0, stream>>>(bufA, (int)U64);
    k_zero<<<cdiv(I64, 256), 256, 0, stream>>>(bufC, (int)I64);
    k_spmm<<<spmmGrid, 256, 0, stream>>>(a_r, a_c, a_v, iE, bufA, NNZ);
    k_spmm<<<spmmGrid, 256, 0, stream>>>(t_r, t_c, t_v, uE, bufC, NNZ);
    k_zero<<<cdiv(U64, 256), 256, 0, stream>>>(bufB, (int)U64);
    k_zero<<<cdiv(I64, 256), 256, 0, stream>>>(bufD, (int)I64);
    k_spmm<<<spmmGrid, 256, 0, stream>>>(a_r, a_c, a_v, bufC, bufB, NNZ);
    k_spmm<<<spmmGrid, 256, 0, stream>>>(t_r, t_c, t_v, bufA, bufD, NNZ);
    k_add3<<<cdiv(U64, 256), 256, 0, stream>>>(o_uiU, uE, bufA, bufB, (int)U64);
    k_add3<<<cdiv(I64, 256), 256, 0, stream>>>(o_uiI, iE, bufC, bufD, (int)I64);

    // -------- lightgcn2 (uu, 2 hops) --------
    k_zero<<<cdiv(U64, 256), 256, 0, stream>>>(bufA, (int)U64);
    k_spmm<<<spmmGrid, 256, 0, stream>>>(u_r, u_c, u_v, uE, bufA, NNZ);
    k_zero<<<cdiv(U64, 256), 256, 0, stream>>>(bufB, (int)U64);
    k_spmm<<<spmmGrid, 256, 0, stream>>>(u_r, u_c, u_v, bufA, bufB, NNZ);
    k_add3<<<cdiv(U64, 256), 256, 0, stream>>>(o_uuE, uE, bufA, bufB, (int)U64);

    // -------- prepare_key: Key = (E @ K) in [H,N,DH] layout (WMMA + TDM) --------
    const int uTiles = U_N / 16, iTiles = I_N / 16;
    k_gemm64<<<cdiv(uTiles, 8), 256, 0, stream>>>(o_uiU, Ks + 0,    o_uiUKey, uTiles, U_N, 1);
    k_gemm64<<<cdiv(iTiles, 8), 256, 0, stream>>>(o_uiI, Ks + 4096, o_uiIKey, iTiles, I_N, 1);
    k_gemm64<<<cdiv(uTiles, 8), 256, 0, stream>>>(o_uuE, Ks + 8192, o_uuKey,  uTiles, U_N, 1);

    // -------- hg_transformer x3 --------
    run_transformer(o_uiU, o_uiUKey, Hyp + 0 * 8192, hgV + 0 * 4096, hgpV + 0 * 4096,
                    hgW1 + 0 * 2 * 16384, hgW2 + 0 * 2 * 16384,
                    o_uiUlat, o_uiUHyp, bufA, bufB, sm, U_N, stream);
    run_transformer(o_uiI, o_uiIKey, Hyp + 1 * 8192, hgV + 1 * 4096, hgpV + 1 * 4096,
                    hgW1 + 1 * 2 * 16384, hgW2 + 1 * 2 * 16384,
                    o_uiIlat, o_uiIHyp, bufA, bufB, sm, I_N, stream);
    run_transformer(o_uuE, o_uuKey, Hyp + 2 * 8192, hgV + 2 * 4096, hgpV + 2 * 4096,
                    hgW1 + 2 * 2 * 16384, hgW2 + 2 * 2 * 16384,
                    o_uuLat, o_uuHyp, bufA, bufB, sm, U_N, stream);
}